// CropAndResize_5669356832751
// MI455X (gfx1250) — compile-verified
//
#include <hip/hip_runtime.h>

typedef __attribute__((ext_vector_type(2))) float v2f;
typedef __attribute__((ext_vector_type(8))) float v8f;

#define CROP_H 14
#define CROP_W 14
#define IMG_C  256
#define IMG_H  64
#define IMG_W  64
#define CBLK   16   // channels per workgroup = WMMA N
#define TPAD   17   // LDS pad for conflict-free access
#define WXPAD  65

// out(16x16: x-by-c) = Wx(16x64) @ T_y(64x16) per y, per (box, 16-channel block).
// y-interp done sparsely (row lerp) into LDS; x-interp as dense WMMA GEMM.
// OOB handling folded into weights (EXTRAP == 0).
__global__ __launch_bounds__(256)
void crop_resize_wmma(const float* __restrict__ img,
                      const float* __restrict__ boxes,
                      const int*   __restrict__ box_ind,
                      float* __restrict__ out, int nBoxes)
{
    __shared__ float sWx[16 * WXPAD];                 // 4160 B
    __shared__ float sT [CROP_H * IMG_W * TPAD];      // 60928 B
    __shared__ float sYl[CROP_H];
    __shared__ int   sTi[CROP_H], sBi[CROP_H], sOy[CROP_H];

    const int tid = threadIdx.x;
    const int n   = blockIdx.x / (IMG_C / CBLK);
    const int c0  = (blockIdx.x % (IMG_C / CBLK)) * CBLK;
    if (n >= nBoxes) return;

    const float y1 = boxes[n*4+0], x1 = boxes[n*4+1];
    const float y2 = boxes[n*4+2], x2 = boxes[n*4+3];
    const int   b  = box_ind[n];
    const float hs = (y2 - y1) * (float)(IMG_H - 1) / (float)(CROP_H - 1);
    const float wsc= (x2 - x1) * (float)(IMG_W - 1) / (float)(CROP_W - 1);

    // --- per-box y coefficients (2-sparse rows) ---
    if (tid < CROP_H) {
        float iny = y1 * (float)(IMG_H - 1) + (float)tid * hs;
        int   oob = (iny < 0.f) || (iny > (float)(IMG_H - 1));
        float tf  = floorf(iny);
        sYl[tid] = iny - tf;
        sTi[tid] = (int)fminf(fmaxf(tf,       0.f), (float)(IMG_H - 1));
        sBi[tid] = (int)fminf(fmaxf(tf + 1.f, 0.f), (float)(IMG_H - 1));
        sOy[tid] = oob;
    }
    // --- build x weight matrix Wx[16][64]; oob_x / pad rows are zero ---
    for (int idx = tid; idx < 16 * IMG_W; idx += 256) {
        int   x = idx / IMG_W, j = idx % IMG_W;
        float w = 0.f;
        if (x < CROP_W) {
            float inx = x1 * (float)(IMG_W - 1) + (float)x * wsc;
            if (!((inx < 0.f) || (inx > (float)(IMG_W - 1)))) {
                float lf = floorf(inx);
                float xl = inx - lf;
                int li = (int)fminf(fmaxf(lf,       0.f), (float)(IMG_W - 1));
                int ri = (int)fminf(fmaxf(lf + 1.f, 0.f), (float)(IMG_W - 1));
                if (j == li) w += 1.f - xl;
                if (j == ri) w += xl;
            }
        }
        sWx[x * WXPAD + j] = w;
    }
    __syncthreads();

    // --- Phase A: sparse y-lerp into LDS: T[y][j][c] (coalesced row loads) ---
    const int imgBase = ((b * IMG_C + c0) * IMG_H) * IMG_W;
    for (int idx = tid; idx < CROP_H * CBLK * IMG_W; idx += 256) {
        int j = idx & (IMG_W - 1);
        int c = (idx >> 6) & (CBLK - 1);
        int y = idx >> 10;
        float v = 0.f;
        if (!sOy[y]) {
            int base = imgBase + c * (IMG_H * IMG_W) + j;
            float rt = img[base + sTi[y] * IMG_W];
            float rb = img[base + sBi[y] * IMG_W];
            v = rt + (rb - rt) * sYl[y];
        }
        sT[(y * IMG_W + j) * TPAD + c] = v;
    }
    __syncthreads();

    // --- Phase B: chained V_WMMA_F32_16X16X4_F32, K = 64 in 16 steps ---
    const int wave = tid >> 5;
    const int lane = tid & 31;
    const int m    = lane & 15;              // A row (x) / B,D col (c)
    const int koff = (lane < 16) ? 0 : 2;    // K slot per half-wave (A/B layout)

    v2f a[16];                               // Wx is y-invariant: hoist A operands
    #pragma unroll
    for (int k = 0; k < 16; ++k) {
        a[k].x = sWx[m * WXPAD + 4*k + koff];
        a[k].y = sWx[m * WXPAD + 4*k + koff + 1];
    }

    for (int y = wave; y < CROP_H; y += 8) { // wave-uniform loop: EXEC stays full
        v8f acc = {};
        #pragma unroll
        for (int k = 0; k < 16; ++k) {
            v2f bm;
            bm.x = sT[(y * IMG_W + 4*k + koff    ) * TPAD + m];
            bm.y = sT[(y * IMG_W + 4*k + koff + 1) * TPAD + m];
            acc = __builtin_amdgcn_wmma_f32_16x16x4_f32(
                      false, a[k], false, bm, (short)0, acc, false, false);
        }
        // D layout: VGPR r -> x = r (lanes 0-15) or r+8 (lanes 16-31), c = lane%16
        size_t outBase = ((size_t)n * IMG_C + c0 + m) * (CROP_H * CROP_W)
                       + (size_t)y * CROP_W;
        int xoff = (lane < 16) ? 0 : 8;
        #pragma unroll
        for (int r = 0; r < 8; ++r) {
            int x = r + xoff;
            if (x < CROP_W) out[outBase + x] = acc[r];
        }
    }
}

extern "C" void kernel_launch(void* const* d_in, const int* in_sizes, int n_in,
                              void* d_out, int out_size, void* d_ws, size_t ws_size,
                              hipStream_t stream) {
    const float* img     = (const float*)d_in[0];
    const float* boxes   = (const float*)d_in[1];
    const int*   box_ind = (const int*)d_in[2];
    float* out = (float*)d_out;
    int nBoxes = in_sizes[2];
    dim3 grid(nBoxes * (IMG_C / CBLK));
    crop_resize_wmma<<<grid, 256, 0, stream>>>(img, boxes, box_ind, out, nBoxes);
}